// QModel_65481071407461
// MI455X (gfx1250) — compile-verified
//
#include <hip/hip_runtime.h>

#define NW      12
#define DIM     4096            // 2^12 amplitudes
#define NLAYERS 2
#define NGATES  (NLAYERS * NW)
#define TPB     256             // 8 wave32 waves

typedef __attribute__((ext_vector_type(2))) float v2f;
typedef __attribute__((ext_vector_type(8))) float v8f;

struct cpx { float re, im; };
__device__ inline cpx cmul(cpx a, cpx b) { return { a.re*b.re - a.im*b.im, a.re*b.im + a.im*b.re }; }
__device__ inline cpx cadd(cpx a, cpx b) { return { a.re + b.re, a.im + b.im }; }

// One workgroup per batch element. Statevector lives in LDS (32 KB).
// Gates applied via V_WMMA_F32_16X16X4_F32: A = 4x4 real gate matrix (padded
// to 16x4), B = 16 amplitude pairs as columns (4x16), D rows 0..3 = new pairs.
__global__ __launch_bounds__(TPB) void qsim_kernel(
    const float* __restrict__ xb,      // (B, 12)   state_batch
    const float* __restrict__ params,  // (2, 12, 3) ansatz_params
    const float* __restrict__ hw,      // (12)      head_w
    const float* __restrict__ hb,      // (1)       head_b
    float* __restrict__ out)           // (B)
{
    __shared__ float2 st[DIM];             // interleaved (re, im)
    __shared__ float  Gm[NGATES * 16];     // per-gate A matrix, 4x4 row-major
    __shared__ float  hc[NW], hs[NW], hww[NW];
    __shared__ float  wsum[TPB / 32];

    const int b    = blockIdx.x;
    const int t    = threadIdx.x;
    const int lane = t & 31;
    const int wv   = t >> 5;

    // --- per-wire init angles + head weights ---
    if (t < NW) {
        float x = 0.5f * xb[b * NW + t];
        hc[t]  = cosf(x);
        hs[t]  = sinf(x);
        hww[t] = hw[t];
    }
    // --- precompute all 24 gate matrices: U = Rz(tz) Ry(ty) Rx(tx) ---
    if (t < NGATES) {
        float tx = params[t * 3 + 0], ty = params[t * 3 + 1], tz = params[t * 3 + 2];
        float cx = cosf(0.5f * tx), sx = sinf(0.5f * tx);
        float cy = cosf(0.5f * ty), sy = sinf(0.5f * ty);
        float cz = cosf(0.5f * tz), sz = sinf(0.5f * tz);
        cpx a00{cx, 0.f}, a01{0.f, -sx}, a10{0.f, -sx}, a11{cx, 0.f};   // Rx
        cpx b00{cy, 0.f}, b01{-sy, 0.f}, b10{sy, 0.f},  b11{cy, 0.f};   // Ry
        cpx m00 = cadd(cmul(b00, a00), cmul(b01, a10));
        cpx m01 = cadd(cmul(b00, a01), cmul(b01, a11));
        cpx m10 = cadd(cmul(b10, a00), cmul(b11, a10));
        cpx m11 = cadd(cmul(b10, a01), cmul(b11, a11));
        cpx z0{cz, -sz}, z1{cz, sz};                                    // Rz diag
        cpx U00 = cmul(z0, m00), U01 = cmul(z0, m01);
        cpx U10 = cmul(z1, m10), U11 = cmul(z1, m11);
        // A[j][k]: coefficient of old component k into new component j
        // (re0', im0', re1', im1') from (re0, im0, re1, im1)
        float* A = &Gm[t * 16];
        A[ 0] = U00.re; A[ 1] = -U00.im; A[ 2] = U01.re; A[ 3] = -U01.im;
        A[ 4] = U00.im; A[ 5] =  U00.re; A[ 6] = U01.im; A[ 7] =  U01.re;
        A[ 8] = U10.re; A[ 9] = -U10.im; A[10] = U11.re; A[11] = -U11.im;
        A[12] = U10.im; A[13] =  U10.re; A[14] = U11.im; A[15] =  U11.re;
    }
    __syncthreads();

    // --- initial product state (wire 0 = MSB): amp(i) = prod_w (bit? sin : cos) ---
    for (int j = 0; j < DIM / TPB; ++j) {
        int idx = t + j * TPB;
        float amp = 1.0f;
#pragma unroll
        for (int w = 0; w < NW; ++w)
            amp *= ((idx >> (NW - 1 - w)) & 1) ? hs[w] : hc[w];
        st[idx] = make_float2(amp, 0.0f);
    }
    __syncthreads();

    // --- circuit ---
    for (int layer = 0; layer < NLAYERS; ++layer) {
        // 12 Rot gates via WMMA
        for (int w = 0; w < NW; ++w) {
            const int g    = layer * NW + w;
            const int p    = NW - 1 - w;       // target bit position (LSB-based)
            const int mask = (1 << p) - 1;
            const int r    = lane & 15;
            // A operand: lanes 0-15 hold K=0,1 / lanes 16-31 hold K=2,3 of row r
            v2f av;
            if (r < 4) {
                int kh = (lane >> 4) << 1;     // 0 or 2
                av[0] = Gm[g * 16 + r * 4 + kh + 0];
                av[1] = Gm[g * 16 + r * 4 + kh + 1];
            } else {
                av[0] = 0.f; av[1] = 0.f;
            }
            // 2048 pairs / 16 per WMMA = 128 chunks, 16 per wave
            for (int c = 0; c < 16; ++c) {
                int pairIdx = ((wv * 16 + c) << 4) | r;
                int i0 = ((pairIdx & ~mask) << 1) | (pairIdx & mask);
                int i1 = i0 | (1 << p);
                // B operand: lane n -> (re0,im0) of pair n; lane n+16 -> (re1,im1)
                float2 sv = st[(lane < 16) ? i0 : i1];
                v2f bv; bv[0] = sv.x; bv[1] = sv.y;
                v8f acc = {};
                v8f d = __builtin_amdgcn_wmma_f32_16x16x4_f32(
                    false, av, false, bv, (short)0, acc, false, false);
                // D: lane n (0-15), VGPRs 0..3 = rows 0..3 = new (re0,im0,re1,im1)
                if (lane < 16) {
                    st[i0] = make_float2(d[0], d[1]);
                    st[i1] = make_float2(d[2], d[3]);
                }
            }
            __syncthreads();
        }
        // CNOT chain c -> c+1: swap amplitudes (ctrl=1, tgt=0) <-> (ctrl=1, tgt=1)
        for (int cc = 0; cc < NW - 1; ++cc) {
            const int pt = NW - 2 - cc;        // target bit; control bit = pt+1
            for (int j = 0; j < (DIM / 4) / TPB; ++j) {   // 1024 swaps, 4 per thread
                int k    = t + j * TPB;
                int low  = k & ((1 << pt) - 1);
                int high = k >> pt;
                int i0   = (high << (pt + 2)) | (1 << (pt + 1)) | low;
                int i1   = i0 | (1 << pt);
                float2 s0 = st[i0];
                float2 s1 = st[i1];
                st[i0] = s1;
                st[i1] = s0;
            }
            __syncthreads();
        }
    }

    // --- fused Z-expectations + head: out = sum_i p_i * sum_w sign_w(i)*hw[w] + hb ---
    float local = 0.f;
    for (int j = 0; j < DIM / TPB; ++j) {
        int idx = t + j * TPB;
        float2 s = st[idx];
        float pr = s.x * s.x + s.y * s.y;
        float gsum = 0.f;
#pragma unroll
        for (int w = 0; w < NW; ++w) {
            float hv = hww[w];
            gsum += ((idx >> (NW - 1 - w)) & 1) ? -hv : hv;
        }
        local += pr * gsum;
    }
#pragma unroll
    for (int off = 16; off >= 1; off >>= 1)
        local += __shfl_xor(local, off, 32);
    if (lane == 0) wsum[wv] = local;
    __syncthreads();
    if (t == 0) {
        float tot = hb[0];
#pragma unroll
        for (int i = 0; i < TPB / 32; ++i) tot += wsum[i];
        out[b] = tot;
    }
}

extern "C" void kernel_launch(void* const* d_in, const int* in_sizes, int n_in,
                              void* d_out, int out_size, void* d_ws, size_t ws_size,
                              hipStream_t stream) {
    const float* xb     = (const float*)d_in[0];   // state_batch (B,12)
    const float* params = (const float*)d_in[1];   // ansatz_params (2,12,3)
    const float* hw     = (const float*)d_in[2];   // head_w (1,12)
    const float* hb     = (const float*)d_in[3];   // head_b (1,)
    float* out = (float*)d_out;
    int B = in_sizes[0] / NW;
    qsim_kernel<<<B, TPB, 0, stream>>>(xb, params, hw, hb, out);
}